// OffsetMPT_22308060136264
// MI455X (gfx1250) — compile-verified
//
#include <hip/hip_runtime.h>
#include <math.h>

typedef __attribute__((ext_vector_type(2))) float v2f;
typedef __attribute__((ext_vector_type(8))) float v8f;

__device__ __forceinline__ float gelu_exact(float v) {
    return 0.5f * v * (1.0f + erff(v * 0.70710678118654752f));
}

// ---------------------------------------------------------------------------
// Encoder: h = (gelu(x@w1+b1) |> LN) @ w2 + b2     (thread per node)
// ---------------------------------------------------------------------------
__global__ __launch_bounds__(256) void encoder_kernel(
    const float* __restrict__ x, const float* __restrict__ w1, const float* __restrict__ b1,
    const float* __restrict__ lg, const float* __restrict__ lb,
    const float* __restrict__ w2, const float* __restrict__ b2,
    float* __restrict__ h, int N_)
{
    __shared__ float sw1[6 * 64];
    __shared__ float sw2[64 * 64];
    __shared__ float sb1[64], sb2[64], slg[64], slb[64];
    for (int i = threadIdx.x; i < 6 * 64; i += blockDim.x) sw1[i] = w1[i];
    for (int i = threadIdx.x; i < 64 * 64; i += blockDim.x) sw2[i] = w2[i];
    if (threadIdx.x < 64) {
        sb1[threadIdx.x] = b1[threadIdx.x];
        sb2[threadIdx.x] = b2[threadIdx.x];
        slg[threadIdx.x] = lg[threadIdx.x];
        slb[threadIdx.x] = lb[threadIdx.x];
    }
    __syncthreads();
    int n = blockIdx.x * blockDim.x + threadIdx.x;
    if (n >= N_) return;

    float xi[6];
    #pragma unroll
    for (int d = 0; d < 6; ++d) xi[d] = x[(size_t)n * 6 + d];

    float h1[64];
    float mean = 0.0f;
    #pragma unroll
    for (int j = 0; j < 64; ++j) {
        float s = sb1[j];
        #pragma unroll
        for (int d = 0; d < 6; ++d) s += xi[d] * sw1[d * 64 + j];
        s = gelu_exact(s);
        h1[j] = s;
        mean += s;
    }
    mean *= (1.0f / 64.0f);
    float var = 0.0f;
    #pragma unroll
    for (int j = 0; j < 64; ++j) { float d = h1[j] - mean; var += d * d; }
    var *= (1.0f / 64.0f);
    float inv = rsqrtf(var + 1e-5f);
    #pragma unroll
    for (int j = 0; j < 64; ++j) h1[j] = (h1[j] - mean) * inv * slg[j] + slb[j];

    float* hp = h + (size_t)n * 64;
    for (int j = 0; j < 64; ++j) {
        float s = sb2[j];
        #pragma unroll
        for (int i = 0; i < 64; ++i) s += h1[i] * sw2[i * 64 + j];
        hp[j] = s;
    }
}

// ---------------------------------------------------------------------------
// Fused q/k/v/skip projection with V_WMMA_F32_16X16X4_F32.
// One wave per 16-node M-tile; K=64 -> 16 chained k=4 WMMA steps per tile.
// A frag (16x4 f32): lanes 0-15 rows M=0..15 K={k,k+1}; lanes 16-31 K={k+2,k+3}
// B frag (4x16 f32): lanes 0-15 cols, rows K={k,k+1}; lanes 16-31 K={k+2,k+3}
// C/D (16x16 f32): vgpr r, lanes 0-15 -> M=r; lanes 16-31 -> M=r+8
// `full` is wave-uniform so the store path is a single branch (EXEC all-1s).
// ---------------------------------------------------------------------------
__device__ __forceinline__ void wmma_tile(const float* __restrict__ W,
                                          const float* __restrict__ bias,
                                          float* __restrict__ out,
                                          int ncols, int n0, int m0, int N_,
                                          bool full, const v2f* a, int lane)
{
    int col   = lane & 15;
    int khalf = lane >> 4;
    float bb = bias[n0 + col];
    v8f acc = {bb, bb, bb, bb, bb, bb, bb, bb};
    const float* wp = W + (size_t)(2 * khalf) * ncols + (n0 + col);
    #pragma unroll
    for (int s = 0; s < 16; ++s) {
        v2f b;
        b.x = wp[(size_t)(4 * s) * ncols];
        b.y = wp[(size_t)(4 * s + 1) * ncols];
        acc = __builtin_amdgcn_wmma_f32_16x16x4_f32(
            false, a[s], false, b, (short)0, acc, false, false);
    }
    float* op = out + (size_t)(m0 + 8 * khalf) * ncols + (n0 + col);
    if (full) {
        // common path: wave-uniform, no exec-mask manipulation
        #pragma unroll
        for (int r = 0; r < 8; ++r) op[(size_t)r * ncols] = acc[r];
    } else {
        #pragma unroll
        for (int r = 0; r < 8; ++r) {
            if (m0 + 8 * khalf + r < N_) op[(size_t)r * ncols] = acc[r];
        }
    }
}

__global__ __launch_bounds__(32) void proj_kernel(
    const float* __restrict__ h, int N_,
    const float* __restrict__ Wq, const float* __restrict__ bq,
    const float* __restrict__ Wk, const float* __restrict__ bk,
    const float* __restrict__ Wv, const float* __restrict__ bv_,
    const float* __restrict__ Ws, const float* __restrict__ bs,
    float* __restrict__ q, float* __restrict__ k, float* __restrict__ v,
    float* __restrict__ xr)
{
    int lane  = threadIdx.x;
    int m0    = blockIdx.x * 16;
    int row   = lane & 15;
    int khalf = lane >> 4;
    bool full = (m0 + 16 <= N_);          // wave-uniform
    int rr = m0 + row; if (rr >= N_) rr = N_ - 1;

    v2f a[16];
    const float* hp = h + (size_t)rr * 64 + 2 * khalf;
    #pragma unroll
    for (int s = 0; s < 16; ++s) { a[s].x = hp[4 * s]; a[s].y = hp[4 * s + 1]; }

    #pragma unroll 1
    for (int n0 = 0; n0 < 256; n0 += 16) {
        wmma_tile(Wq, bq,  q, 256, n0, m0, N_, full, a, lane);
        wmma_tile(Wk, bk,  k, 256, n0, m0, N_, full, a, lane);
        wmma_tile(Wv, bv_, v, 256, n0, m0, N_, full, a, lane);
    }
    #pragma unroll 1
    for (int n0 = 0; n0 < 64; n0 += 16)
        wmma_tile(Ws, bs, xr, 64, n0, m0, N_, full, a, lane);
}

// ---------------------------------------------------------------------------
// Edge pass A: alpha[e,h] = dot(q[dst,h], k[src,h]+e[h]) / 8 ; atomic-max per dst
// One wave per edge; 8 lanes per head (8 channels each), shfl_xor reduce.
// Float max via monotone uint mapping.
// ---------------------------------------------------------------------------
__global__ __launch_bounds__(256) void edge_alpha_kernel(
    const int* __restrict__ ei, const float* __restrict__ ea,
    const float* __restrict__ We, const float* __restrict__ q,
    const float* __restrict__ k, float* __restrict__ alpha,
    unsigned* __restrict__ mmax, int E_)
{
    int gid  = blockIdx.x * blockDim.x + threadIdx.x;
    int e    = gid >> 5;
    int lane = threadIdx.x & 31;
    if (e >= E_) return;
    int src = ei[e];
    int dst = ei[E_ + e];
    int hh = lane >> 3;
    int c0 = (lane & 7) << 3;
    int hc = hh * 64 + c0;
    float ea0 = ea[(size_t)e * 4 + 0], ea1 = ea[(size_t)e * 4 + 1];
    float ea2 = ea[(size_t)e * 4 + 2], ea3 = ea[(size_t)e * 4 + 3];
    const float* qp = q + (size_t)dst * 256 + hc;
    const float* kp = k + (size_t)src * 256 + hc;
    const float* wp = We + hc;
    float part = 0.0f;
    #pragma unroll
    for (int c = 0; c < 8; ++c) {
        float ec = ea0 * wp[c] + ea1 * wp[256 + c] + ea2 * wp[512 + c] + ea3 * wp[768 + c];
        part += qp[c] * (kp[c] + ec);
    }
    part += __shfl_xor(part, 1, 32);
    part += __shfl_xor(part, 2, 32);
    part += __shfl_xor(part, 4, 32);
    if ((lane & 7) == 0) {
        float al = part * 0.125f;                // 1/sqrt(C), C=64
        alpha[(size_t)e * 4 + hh] = al;
        unsigned bits = __float_as_uint(al);
        unsigned u = (bits & 0x80000000u) ? ~bits : (bits | 0x80000000u);
        atomicMax(mmax + (size_t)dst * 4 + hh, u);
    }
}

// ---------------------------------------------------------------------------
// Edge pass B: ex = exp(alpha - m[dst]); alpha <- ex; atomicAdd sum[dst]
// Thread per (edge, head).
// ---------------------------------------------------------------------------
__global__ __launch_bounds__(256) void edge_softmax_kernel(
    const int* __restrict__ ei, float* __restrict__ alpha,
    const unsigned* __restrict__ mmax, float* __restrict__ ssum, int E_)
{
    int t = blockIdx.x * blockDim.x + threadIdx.x;
    if (t >= E_ * 4) return;
    int e  = t >> 2;
    int hh = t & 3;
    int dst = ei[E_ + e];
    unsigned um = mmax[(size_t)dst * 4 + hh];
    float m = (um & 0x80000000u) ? __uint_as_float(um & 0x7FFFFFFFu)
                                 : __uint_as_float(~um);
    float ex_ = expf(alpha[t] - m);
    alpha[t] = ex_;
    atomicAdd(ssum + (size_t)dst * 4 + hh, ex_);
}

// ---------------------------------------------------------------------------
// Edge pass C: oacc[dst] += (v[src]+e) * attn   (f32 L2 atomics)
// ---------------------------------------------------------------------------
__global__ __launch_bounds__(256) void edge_msg_kernel(
    const int* __restrict__ ei, const float* __restrict__ ea,
    const float* __restrict__ We, const float* __restrict__ v,
    const float* __restrict__ exv, const float* __restrict__ ssum,
    float* __restrict__ oacc, int E_)
{
    int gid  = blockIdx.x * blockDim.x + threadIdx.x;
    int e    = gid >> 5;
    int lane = threadIdx.x & 31;
    if (e >= E_) return;
    int src = ei[e];
    int dst = ei[E_ + e];
    int hh = lane >> 3;
    int c0 = (lane & 7) << 3;
    int hc = hh * 64 + c0;
    float ea0 = ea[(size_t)e * 4 + 0], ea1 = ea[(size_t)e * 4 + 1];
    float ea2 = ea[(size_t)e * 4 + 2], ea3 = ea[(size_t)e * 4 + 3];
    float attn = exv[(size_t)e * 4 + hh] / (ssum[(size_t)dst * 4 + hh] + 1e-16f);
    const float* vp = v + (size_t)src * 256 + hc;
    const float* wp = We + hc;
    float* op = oacc + (size_t)dst * 256 + hc;
    #pragma unroll
    for (int c = 0; c < 8; ++c) {
        float ec = ea0 * wp[c] + ea1 * wp[256 + c] + ea2 * wp[512 + c] + ea3 * wp[768 + c];
        atomicAdd(op + c, (vp[c] + ec) * attn);
    }
}

// ---------------------------------------------------------------------------
// Node update: head-mean, beta gate, residual, LayerNorm  (thread per node)
// ---------------------------------------------------------------------------
__global__ __launch_bounds__(256) void node_update_kernel(
    float* __restrict__ h, const float* __restrict__ oacc,
    const float* __restrict__ xrg, const float* __restrict__ Wb,
    const float* __restrict__ lng, const float* __restrict__ lnb, int N_)
{
    int n = blockIdx.x * blockDim.x + threadIdx.x;
    if (n >= N_) return;
    const float* op = oacc + (size_t)n * 256;
    const float* xp = xrg + (size_t)n * 64;
    float* hp = h + (size_t)n * 64;

    float o[64];
    float lgt = 0.0f;
    #pragma unroll
    for (int c = 0; c < 64; ++c) {
        float oc = 0.25f * (op[c] + op[64 + c] + op[128 + c] + op[192 + c]);
        o[c] = oc;
        float xc = xp[c];
        lgt += oc * Wb[c] + xc * Wb[64 + c] + (oc - xc) * Wb[128 + c];
    }
    float beta = 1.0f / (1.0f + expf(-lgt));

    float mean = 0.0f;
    #pragma unroll
    for (int c = 0; c < 64; ++c) {
        float xc = xp[c];
        float hn = beta * xc + (1.0f - beta) * o[c];
        float t = hp[c] + hn;
        o[c] = t;
        mean += t;
    }
    mean *= (1.0f / 64.0f);
    float var = 0.0f;
    #pragma unroll
    for (int c = 0; c < 64; ++c) { float d = o[c] - mean; var += d * d; }
    var *= (1.0f / 64.0f);
    float inv = rsqrtf(var + 1e-5f);
    #pragma unroll
    for (int c = 0; c < 64; ++c)
        hp[c] = (o[c] - mean) * inv * lng[c] + lnb[c];
}

// ---------------------------------------------------------------------------
// Output head: out = gelu(h@w1+b1) @ w2 + b2     (thread per node)
// ---------------------------------------------------------------------------
__global__ __launch_bounds__(256) void head_kernel(
    const float* __restrict__ h, const float* __restrict__ w1,
    const float* __restrict__ b1, const float* __restrict__ w2,
    const float* __restrict__ b2, float* __restrict__ out, int N_)
{
    __shared__ float sw1[64 * 32];
    __shared__ float sw2[64];
    __shared__ float sb1[32];
    for (int i = threadIdx.x; i < 64 * 32; i += blockDim.x) sw1[i] = w1[i];
    if (threadIdx.x < 64) sw2[threadIdx.x] = w2[threadIdx.x];
    if (threadIdx.x < 32) sb1[threadIdx.x] = b1[threadIdx.x];
    __syncthreads();
    int n = blockIdx.x * blockDim.x + threadIdx.x;
    if (n >= N_) return;

    float hv[64];
    const float* hp = h + (size_t)n * 64;
    #pragma unroll
    for (int i = 0; i < 64; ++i) hv[i] = hp[i];
    float a0 = b2[0], a1 = b2[1];
    for (int j = 0; j < 32; ++j) {
        float s = sb1[j];
        #pragma unroll
        for (int i = 0; i < 64; ++i) s += hv[i] * sw1[i * 32 + j];
        s = gelu_exact(s);
        a0 += s * sw2[j * 2];
        a1 += s * sw2[j * 2 + 1];
    }
    out[(size_t)n * 2]     = a0;
    out[(size_t)n * 2 + 1] = a1;
}

// ---------------------------------------------------------------------------
extern "C" void kernel_launch(void* const* d_in, const int* in_sizes, int n_in,
                              void* d_out, int out_size, void* d_ws, size_t ws_size,
                              hipStream_t stream)
{
    (void)n_in; (void)out_size; (void)ws_size;
    const float* x      = (const float*)d_in[0];
    const int*   ei     = (const int*)d_in[1];   // edge_index [2,E]
    const float* ea     = (const float*)d_in[2];
    const float* enc_w1 = (const float*)d_in[3];
    const float* enc_b1 = (const float*)d_in[4];
    const float* enc_lg = (const float*)d_in[5];
    const float* enc_lb = (const float*)d_in[6];
    const float* enc_w2 = (const float*)d_in[7];
    const float* enc_b2 = (const float*)d_in[8];
    const float* Wq  = (const float*)d_in[9];
    const float* bq  = (const float*)d_in[10];
    const float* Wk  = (const float*)d_in[11];
    const float* bk  = (const float*)d_in[12];
    const float* Wv  = (const float*)d_in[13];
    const float* bv  = (const float*)d_in[14];
    const float* We  = (const float*)d_in[15];
    const float* Wsk = (const float*)d_in[16];
    const float* bsk = (const float*)d_in[17];
    const float* Wb  = (const float*)d_in[18];
    const float* lng = (const float*)d_in[19];
    const float* lnb = (const float*)d_in[20];
    const float* hw1 = (const float*)d_in[21];
    const float* hb1 = (const float*)d_in[22];
    const float* hw2 = (const float*)d_in[23];
    const float* hb2 = (const float*)d_in[24];

    const int N = in_sizes[0] / 6;
    const int E = in_sizes[1] / 2;
    const int L = 2;

    char* p = (char*)d_ws;
    float*    h     = (float*)p;    p += (size_t)N * 64 * 4;
    float*    q     = (float*)p;    p += (size_t)N * 256 * 4;
    float*    k     = (float*)p;    p += (size_t)N * 256 * 4;
    float*    v     = (float*)p;    p += (size_t)N * 256 * 4;
    float*    xr    = (float*)p;    p += (size_t)N * 64 * 4;
    float*    alpha = (float*)p;    p += (size_t)E * 4 * 4;
    unsigned* mmax  = (unsigned*)p; p += (size_t)N * 4 * 4;
    float*    ssum  = (float*)p;    p += (size_t)N * 4 * 4;
    float*    oacc  = (float*)p;    p += (size_t)N * 256 * 4;

    encoder_kernel<<<(N + 255) / 256, 256, 0, stream>>>(
        x, enc_w1, enc_b1, enc_lg, enc_lb, enc_w2, enc_b2, h, N);

    for (int l = 0; l < L; ++l) {
        proj_kernel<<<(N + 15) / 16, 32, 0, stream>>>(
            h, N,
            Wq  + (size_t)l * 64 * 256, bq  + (size_t)l * 256,
            Wk  + (size_t)l * 64 * 256, bk  + (size_t)l * 256,
            Wv  + (size_t)l * 64 * 256, bv  + (size_t)l * 256,
            Wsk + (size_t)l * 64 * 64,  bsk + (size_t)l * 64,
            q, k, v, xr);

        hipMemsetAsync(mmax, 0, (size_t)N * 4 * 4, stream);
        hipMemsetAsync(ssum, 0, (size_t)N * 4 * 4, stream);
        hipMemsetAsync(oacc, 0, (size_t)N * 256 * 4, stream);

        const float* Wel = We + (size_t)l * 4 * 256;
        edge_alpha_kernel<<<(E + 7) / 8, 256, 0, stream>>>(
            ei, ea, Wel, q, k, alpha, mmax, E);
        edge_softmax_kernel<<<((size_t)E * 4 + 255) / 256, 256, 0, stream>>>(
            ei, alpha, mmax, ssum, E);
        edge_msg_kernel<<<(E + 7) / 8, 256, 0, stream>>>(
            ei, ea, Wel, v, alpha, ssum, oacc, E);
        node_update_kernel<<<(N + 255) / 256, 256, 0, stream>>>(
            h, oacc, xr, Wb + (size_t)l * 192, lng + (size_t)l * 64,
            lnb + (size_t)l * 64, N);
    }

    head_kernel<<<(N + 255) / 256, 256, 0, stream>>>(
        h, hw1, hb1, hw2, hb2, (float*)d_out, N);
}